// TreeLogicLayer_41223096107631
// MI455X (gfx1250) — compile-verified
//
#include <hip/hip_runtime.h>

// ---------------------------------------------------------------------------
// MI455X (gfx1250) implementation.
//
// Roofline: 128 MiB read (x) + 64 MiB write (out) = 192 MiB @ 23.3 TB/s
// => ~8.6 us floor. Compute is ~0.2 GFLOP -> irrelevant. The kernel is a
// pure HBM streamer; the only real matrix op (the coefficient einsum,
// (256x16)@(16x4)) is done natively with V_WMMA_F32_16X16X4_F32.
// ---------------------------------------------------------------------------

typedef __attribute__((ext_vector_type(2))) float v2f;
typedef __attribute__((ext_vector_type(8))) float v8f;

// _GATE_COEFFS[g][k]  (contraction dim g = 16, output dim k = 4)
__device__ __constant__ float GATE[16][4] = {
    {0, 0, 0, 0},  {0, 0, 0, 1},  {0, 1, 0, -1}, {0, 1, 0, 0},
    {0, 0, 1, -1}, {0, 0, 1, 0},  {0, 1, 1, -2}, {0, 1, 1, -1},
    {1, -1, -1, 1},{1, -1, -1, 2},{1, 0, -1, 0}, {1, 0, -1, 1},
    {1, -1, 0, 0}, {1, -1, 0, 1}, {1, 0, 0, -1}, {1, 0, 0, 0}};

// ---------------------------------------------------------------------------
// Kernel 1: coeffs[c][o][k] = softmax(weights[c][o][:]) @ GATE
// One block, 512 threads = 16 wave32's. Phase 1: 256 threads compute one
// softmax row each into LDS. Phase 2: each wave computes a 16-row M-tile of
// the (256x16)@(16x4->padded 16) GEMM with 4 chained f32 WMMAs (K=4 each).
// ---------------------------------------------------------------------------
__global__ __launch_bounds__(512) void coeff_wmma_kernel(
    const float* __restrict__ w, float* __restrict__ coeffs) {
  __shared__ float wprob[256][16];

  const int t = threadIdx.x;
  if (t < 256) {
    // One softmax row (16 contiguous floats) per thread.
    const float4* row = (const float4*)(w + t * 16);
    float v[16];
    float4 r;
    r = row[0]; v[0]=r.x; v[1]=r.y; v[2]=r.z; v[3]=r.w;
    r = row[1]; v[4]=r.x; v[5]=r.y; v[6]=r.z; v[7]=r.w;
    r = row[2]; v[8]=r.x; v[9]=r.y; v[10]=r.z; v[11]=r.w;
    r = row[3]; v[12]=r.x; v[13]=r.y; v[14]=r.z; v[15]=r.w;
    float m = v[0];
#pragma unroll
    for (int i = 1; i < 16; ++i) m = fmaxf(m, v[i]);
    float s = 0.f;
#pragma unroll
    for (int i = 0; i < 16; ++i) { v[i] = __expf(v[i] - m); s += v[i]; }
    const float inv = 1.0f / s;
#pragma unroll
    for (int i = 0; i < 16; ++i) wprob[t][i] = v[i] * inv;
  }
  __syncthreads();

  // WMMA phase: EXEC all-1s (all 512 threads, uniform control flow).
  const int wave = t >> 5;       // 0..15 -> M-tile (rows 16*wave .. +15)
  const int lane = t & 31;
  const int n    = lane & 15;    // N column within tile
  const int hi   = lane >> 4;    // lane half selects K sub-pair / M+8 half
  const int mrow = (wave << 4) + n;  // A fragment row = lane%16

  v8f acc = {};  // zero C
#pragma unroll
  for (int q = 0; q < 4; ++q) {   // K = 16 as 4 chained K=4 steps
    const int g0 = 4 * q + 2 * hi;
    // A: 16x4 f32 fragment. v0 = K=g0 (lanes 0-15) / K=g0 (=4q+2, lanes 16-31)
    v2f a;
    a.x = wprob[mrow][g0];
    a.y = wprob[mrow][g0 + 1];
    // B: 4x16 f32 fragment, N columns striped over lanes; pad N>=4 with 0.
    v2f b;
    b.x = (n < 4) ? GATE[g0][n] : 0.0f;
    b.y = (n < 4) ? GATE[g0 + 1][n] : 0.0f;
    acc = __builtin_amdgcn_wmma_f32_16x16x4_f32(
        /*neg_a=*/false, a, /*neg_b=*/false, b,
        /*c_mod=*/(short)0, acc, /*reuse_a=*/false, /*reuse_b=*/false);
  }

  // D layout: VGPR r -> M=r (lanes 0-15), M=r+8 (lanes 16-31); N = lane%16.
  if (n < 4) {
    const int mbase = (wave << 4) + (hi << 3);
#pragma unroll
    for (int r = 0; r < 8; ++r)
      coeffs[(mbase + r) * 4 + n] = acc[r];
  }
}

// ---------------------------------------------------------------------------
// Kernel 2: bandwidth-bound streamer.
// out layout: ((b*64 + c)*1024 + h*32 + w)*4 + o ; input pair for output
// index i sits at x[2*i], x[2*i+1]. Each thread: 2x float4 in, 1x float4 out.
// c = (out_base >> 12) & 63 = (tid >> 10) & 63 since out_base = tid*4.
// Coefficient table (4 KB) staged in LDS; all lanes of a wave share one c
// -> LDS broadcast reads, no bank conflicts.
// ---------------------------------------------------------------------------
__global__ __launch_bounds__(256) void gate_stream_kernel(
    const float4* __restrict__ x4, const float4* __restrict__ coeffs4,
    float4* __restrict__ out4, int n_out4) {
  __shared__ float4 scoef[256];
  scoef[threadIdx.x] = coeffs4[threadIdx.x];  // 256 rows of (k0,k1,k2,k3)
  __syncthreads();

  const int t = blockIdx.x * blockDim.x + threadIdx.x;
  if (t >= n_out4) return;

  const int c = (t >> 10) & 63;

  const float4 p0 = x4[2 * t];      // x[..., 0..3]
  const float4 p1 = x4[2 * t + 1];  // x[..., 4..7]

  const float4 k0 = scoef[c * 4 + 0];
  const float4 k1 = scoef[c * 4 + 1];
  const float4 k2 = scoef[c * 4 + 2];
  const float4 k3 = scoef[c * 4 + 3];

  // r = (k0 + k1*a) + b*(k2 + k3*a)
  float4 r;
  r.x = fmaf(p0.y, fmaf(p0.x, k0.w, k0.z), fmaf(p0.x, k0.y, k0.x));
  r.y = fmaf(p0.w, fmaf(p0.z, k1.w, k1.z), fmaf(p0.z, k1.y, k1.x));
  r.z = fmaf(p1.y, fmaf(p1.x, k2.w, k2.z), fmaf(p1.x, k2.y, k2.x));
  r.w = fmaf(p1.w, fmaf(p1.z, k3.w, k3.z), fmaf(p1.z, k3.y, k3.x));

  out4[t] = r;
}

extern "C" void kernel_launch(void* const* d_in, const int* in_sizes, int n_in,
                              void* d_out, int out_size, void* d_ws, size_t ws_size,
                              hipStream_t stream) {
  const float* x = (const float*)d_in[0];        // (64,64,32,32,8) f32
  const float* w = (const float*)d_in[1];        // (64,4,16) f32
  float* out = (float*)d_out;                    // (64,64,32,32,4) f32
  float* coeffs = (float*)d_ws;                  // 1024 f32 scratch (4 KB)

  // 1) tiny WMMA einsum: softmax(weights) @ GATE -> coeffs
  coeff_wmma_kernel<<<1, 512, 0, stream>>>(w, coeffs);

  // 2) HBM streamer
  const int n4 = out_size / 4;                   // 4,194,304 float4 outputs
  const int block = 256;
  const int grid = (n4 + block - 1) / block;     // 16384 blocks
  gate_stream_kernel<<<grid, block, 0, stream>>>(
      (const float4*)x, (const float4*)coeffs, (float4*)out, n4);
}